// HGSELLayer_49855980372022
// MI455X (gfx1250) — compile-verified
//
#include <hip/hip_runtime.h>
#include <hip/hip_bf16.h>
#include <math.h>

#define TOKENS 8192
#define DMODEL 512
#define DFF    2048
#define NEXP   64
#define KACT   2
#define CAP    320   // ceil(8192*2/64*1.25)

typedef float v2f __attribute__((ext_vector_type(2)));
typedef float v8f __attribute__((ext_vector_type(8)));
typedef int   v4i __attribute__((ext_vector_type(4)));

// ---- CDNA5 async global->LDS staging (guarded; fallback = reg staging) ----
#if defined(__AMDGCN__) && defined(__has_builtin)
#if __has_builtin(__builtin_amdgcn_global_load_async_to_lds_b128) && \
    __has_builtin(__builtin_amdgcn_s_wait_asynccnt)
#define ASYNC_STAGE 1
#endif
#endif
#ifndef ASYNC_STAGE
#define ASYNC_STAGE 0
#endif

typedef __attribute__((address_space(1))) v4i g1v4i;   // global 128-bit vec
typedef __attribute__((address_space(3))) v4i l3v4i;   // LDS    128-bit vec

// Low 32 bits of a generic shared-memory pointer are the LDS byte address
// build AS pointers via integer casts.
__device__ __forceinline__ l3v4i* to_lds(const void* p) {
    return (l3v4i*)(unsigned long long)(unsigned int)(unsigned long long)p;
}
__device__ __forceinline__ g1v4i* to_glb(const void* p) {
    return (g1v4i*)(unsigned long long)p;
}

// ------------------------------------------------------------------
// Kernel 1: hash router. One wave per token; only hashes 0..1 matter.
// ------------------------------------------------------------------
__global__ __launch_bounds__(256)
void router_kernel(const float* __restrict__ x,
                   const float* __restrict__ wh,
                   int* __restrict__ sel)
{
    const int wave = threadIdx.x >> 5;
    const int lane = threadIdx.x & 31;
    const int t = blockIdx.x * 8 + wave;
    if (t >= TOKENS) return;

    const float* xr = x + (size_t)t * DMODEL;
    float s0 = 0.f, s1 = 0.f;
    for (int d = lane; d < DMODEL; d += 32) {
        const float xv = xr[d];
        s0 = fmaf(xv, wh[d], s0);
        s1 = fmaf(xv, wh[DMODEL + d], s1);
    }
    for (int off = 16; off > 0; off >>= 1) {
        s0 += __shfl_xor(s0, off, 32);
        s1 += __shfl_xor(s1, off, 32);
    }
    if (lane == 0) {
        int b0 = (int)floorf(s0 * 8.0f);   // BUCKET_SCALE=8, SALT=0, LAYER_ID=0
        int b1 = (int)floorf(s1 * 8.0f);
        b0 = ((b0 % NEXP) + NEXP) % NEXP;  // jnp.mod semantics
        b1 = ((b1 % NEXP) + NEXP) % NEXP;
        sel[t * 2 + 0] = b0;
        sel[t * 2 + 1] = b1;
    }
}

// ------------------------------------------------------------------
// Kernel 2: ordered capacity scan (replays reference cumcount order).
// ------------------------------------------------------------------
__global__ __launch_bounds__(64)
void route_scan_kernel(const int* __restrict__ sel,
                       int* __restrict__ pairSlot,   // [TOKENS*KACT]
                       int* __restrict__ slotTok)    // [NEXP*CAP]
{
    const int e = threadIdx.x;   // 64 threads, expert id
    for (int c = 0; c < CAP; ++c) slotTok[e * CAP + c] = -1;
    __syncthreads();

    int cnt = 0;
    for (int i = 0; i < TOKENS * KACT; ++i) {
        const int ei = sel[i];
        if (ei == e) {
            if (cnt < CAP) {
                pairSlot[i] = e * CAP + cnt;
                slotTok[e * CAP + cnt] = i >> 1;
            } else {
                pairSlot[i] = -1;   // dropped (over capacity)
            }
            ++cnt;
        }
    }
}

// ------------------------------------------------------------------
// Kernel 3: pack buf[e][c][:] = x[tok] (zeros for empty slots).
// ------------------------------------------------------------------
__global__ __launch_bounds__(128)
void pack_kernel(const float* __restrict__ x,
                 const int* __restrict__ slotTok,
                 float* __restrict__ buf)
{
    const int slot = blockIdx.x;            // NEXP*CAP blocks
    const int tok  = slotTok[slot];
    float4* dst = (float4*)(buf + (size_t)slot * DMODEL);
    if (tok >= 0) {
        const float4* src = (const float4*)(x + (size_t)tok * DMODEL);
        dst[threadIdx.x] = src[threadIdx.x];
    } else {
        dst[threadIdx.x] = make_float4(0.f, 0.f, 0.f, 0.f);
    }
}

// ------------------------------------------------------------------
// Kernels 4/5: batched fp32 WMMA GEMM, double-buffered LDS staging.
//   Out[e] = act(A[e](MxK) @ B[e](KxN) + bias[e])
// Block = 128 threads = 4 waves; tile 64x64, BK=16; wave owns 32x32.
// Staging: async global->LDS (ASYNCcnt) when available, else regs.
// ------------------------------------------------------------------
template<bool GELU>
__global__ __launch_bounds__(128)
void moe_gemm_wmma(const float* __restrict__ A,    // [E, M, K]
                   const float* __restrict__ Bw,   // [E, K, N]
                   const float* __restrict__ bias, // [E, N]
                   float* __restrict__ Out,        // [E, M, N]
                   int M, int N, int K)
{
    // pad 16->20 floats: row stride 80B (16B aligned for b128 writes),
    // conflict-free float2 fragment reads (gcd(20,64)=4, 16 distinct banks)
    __shared__ float As[2][64][20];
    __shared__ float Bs[2][16][64];

    const int mTiles = M >> 6;
    const int nTiles = N >> 6;
    const int e  = blockIdx.x / (mTiles * nTiles);
    const int r  = blockIdx.x % (mTiles * nTiles);
    const int m0 = (r / nTiles) << 6;
    const int n0 = (r % nTiles) << 6;

    const float* Ae = A  + (size_t)e * M * K;
    const float* Be = Bw + (size_t)e * K * N;

    const int tid  = threadIdx.x;
    const int wave = tid >> 5;
    const int lane = tid & 31;
    const int wm   = (wave >> 1) * 32;
    const int wn   = (wave &  1) * 32;

    const v8f vz = {0.f,0.f,0.f,0.f,0.f,0.f,0.f,0.f};
    v8f acc[2][2] = {{vz, vz}, {vz, vz}};

    const int mrow = lane & 15;
    const int kb0  = (lane >> 4) << 1;   // lanes 0-15: K 0,1 ; 16-31: K 2,3

#if ASYNC_STAGE
    // issue async global->LDS copies for one 64x16 A tile + 16x64 B tile
    auto stage = [&](int sb, int k0) {
        #pragma unroll
        for (int it = 0; it < 2; ++it) {
            const int f  = tid + (it << 7);
            const int ar = f >> 2, ac = (f & 3)  << 2;
            const int br = f >> 4, bc = (f & 15) << 2;
            __builtin_amdgcn_global_load_async_to_lds_b128(
                to_glb(Ae + (size_t)(m0 + ar) * K + k0 + ac),
                to_lds(&As[sb][ar][ac]), 0, 0);
            __builtin_amdgcn_global_load_async_to_lds_b128(
                to_glb(Be + (size_t)(k0 + br) * N + n0 + bc),
                to_lds(&Bs[sb][br][bc]), 0, 0);
        }
    };
#else
    float4 ra[2], rb[2];
    auto loadRegs = [&](int k0) {
        #pragma unroll
        for (int it = 0; it < 2; ++it) {
            const int f  = tid + (it << 7);
            const int ar = f >> 2, ac = (f & 3)  << 2;
            const int br = f >> 4, bc = (f & 15) << 2;
            ra[it] = *(const float4*)(Ae + (size_t)(m0 + ar) * K + k0 + ac);
            rb[it] = *(const float4*)(Be + (size_t)(k0 + br) * N + n0 + bc);
        }
    };
    auto storeLds = [&](int sb) {
        #pragma unroll
        for (int it = 0; it < 2; ++it) {
            const int f  = tid + (it << 7);
            const int ar = f >> 2, ac = (f & 3)  << 2;
            const int br = f >> 4, bc = (f & 15) << 2;
            *(float4*)&As[sb][ar][ac] = ra[it];   // 80B row stride -> aligned
            *(float4*)&Bs[sb][br][bc] = rb[it];
        }
    };
#endif

    auto compute = [&](int sb) {
        #pragma unroll
        for (int kk = 0; kk < 16; kk += 4) {
            v2f a[2], b[2];
            #pragma unroll
            for (int mi = 0; mi < 2; ++mi)
                a[mi] = *(const v2f*)&As[sb][wm + (mi << 4) + mrow][kk + kb0];
            #pragma unroll
            for (int ni = 0; ni < 2; ++ni) {
                const int col = wn + (ni << 4) + mrow;
                b[ni].x = Bs[sb][kk + kb0 + 0][col];
                b[ni].y = Bs[sb][kk + kb0 + 1][col];
            }
            #pragma unroll
            for (int mi = 0; mi < 2; ++mi)
                #pragma unroll
                for (int ni = 0; ni < 2; ++ni)
                    acc[mi][ni] = __builtin_amdgcn_wmma_f32_16x16x4_f32(
                        false, a[mi], false, b[ni],
                        (short)0, acc[mi][ni], false, false);
        }
    };

    // ---- software-pipelined main loop (double-buffered LDS) ----
#if ASYNC_STAGE
    stage(0, 0);
    __builtin_amdgcn_s_wait_asynccnt(0);
    __syncthreads();
    int cur = 0;
    for (int k0 = 0; k0 < K; k0 += 16) {
        const bool more = (k0 + 16) < K;
        if (more) stage(cur ^ 1, k0 + 16);   // overlaps with compute below
        compute(cur);
        if (more) {
            __builtin_amdgcn_s_wait_asynccnt(0);
            __syncthreads();
        }
        cur ^= 1;
    }
#else
    loadRegs(0);
    storeLds(0);
    __syncthreads();
    int cur = 0;
    for (int k0 = 0; k0 < K; k0 += 16) {
        const bool more = (k0 + 16) < K;
        if (more) {
            loadRegs(k0 + 16);               // global loads in flight
            __builtin_prefetch(Ae + (size_t)(m0 + (tid >> 1)) * K + k0 + 32, 0, 0);
        }
        compute(cur);                        // overlaps with loads
        if (more) {
            storeLds(cur ^ 1);
            __syncthreads();
        }
        cur ^= 1;
    }
#endif

    // ---- epilogue: bias (+ exact GELU); C/D layout rows r / r+8 ----
    const int col_l = lane & 15;
    const int rhalf = (lane >> 4) << 3;
    #pragma unroll
    for (int mi = 0; mi < 2; ++mi) {
        #pragma unroll
        for (int ni = 0; ni < 2; ++ni) {
            const int gcol = n0 + wn + (ni << 4) + col_l;
            const float bv = bias[(size_t)e * N + gcol];
            float* obase = Out + (size_t)e * M * N
                               + (size_t)(m0 + wm + (mi << 4) + rhalf) * N + gcol;
            #pragma unroll
            for (int rr = 0; rr < 8; ++rr) {
                float v = acc[mi][ni][rr] + bv;
                if (GELU)
                    v = 0.5f * v * (1.0f + erff(v * 0.70710678118654752f));
                obase[(size_t)rr * N] = v;
            }
        }
    }
}

// ------------------------------------------------------------------
// Kernel 6: combine. out[t] = mean over the 2 routes (dropped -> 0).
// ------------------------------------------------------------------
__global__ __launch_bounds__(128)
void combine_kernel(const float* __restrict__ y,
                    const int* __restrict__ pairSlot,
                    float* __restrict__ out)
{
    const int t  = blockIdx.x;
    const int s0 = pairSlot[t * 2 + 0];
    const int s1 = pairSlot[t * 2 + 1];
    const int j  = threadIdx.x;

    float rx = 0.f, ry = 0.f, rz = 0.f, rw = 0.f;
    if (s0 >= 0) {
        const float4 v = ((const float4*)(y + (size_t)s0 * DMODEL))[j];
        rx += v.x; ry += v.y; rz += v.z; rw += v.w;
    }
    if (s1 >= 0) {
        const float4 v = ((const float4*)(y + (size_t)s1 * DMODEL))[j];
        rx += v.x; ry += v.y; rz += v.z; rw += v.w;
    }
    ((float4*)(out + (size_t)t * DMODEL))[j] =
        make_float4(0.5f * rx, 0.5f * ry, 0.5f * rz, 0.5f * rw);
}

// ------------------------------------------------------------------
extern "C" void kernel_launch(void* const* d_in, const int* in_sizes, int n_in,
                              void* d_out, int out_size, void* d_ws, size_t ws_size,
                              hipStream_t stream)
{
    (void)in_sizes; (void)n_in; (void)out_size; (void)ws_size;

    const float* x  = (const float*)d_in[0];   // [T, 512]
    const float* wh = (const float*)d_in[1];   // [4, 512]
    const float* W1 = (const float*)d_in[2];   // [64, 512, 2048]
    const float* b1 = (const float*)d_in[3];   // [64, 2048]
    const float* W2 = (const float*)d_in[4];   // [64, 2048, 512]
    const float* b2 = (const float*)d_in[5];   // [64, 512]
    float* out = (float*)d_out;                // [T, 512]

    char*  ws  = (char*)d_ws;
    size_t off = 0;
    auto carve = [&](size_t bytes) -> void* {
        void* p = ws + off;
        off += (bytes + 255) & ~(size_t)255;
        return p;
    };
    int*   sel      = (int*)  carve((size_t)TOKENS * KACT * sizeof(int));
    int*   pairSlot = (int*)  carve((size_t)TOKENS * KACT * sizeof(int));
    int*   slotTok  = (int*)  carve((size_t)NEXP * CAP * sizeof(int));
    float* buf      = (float*)carve((size_t)NEXP * CAP * DMODEL * sizeof(float));
    float* h        = (float*)carve((size_t)NEXP * CAP * DFF    * sizeof(float));
    float* yb       = (float*)carve((size_t)NEXP * CAP * DMODEL * sizeof(float));

    router_kernel<<<TOKENS / 8, 256, 0, stream>>>(x, wh, sel);
    route_scan_kernel<<<1, 64, 0, stream>>>(sel, pairSlot, slotTok);
    pack_kernel<<<NEXP * CAP, 128, 0, stream>>>(x, slotTok, buf);

    {   // h = gelu(buf @ W1 + b1)   M=320, N=2048, K=512
        const int grid = NEXP * (CAP / 64) * (DFF / 64);
        moe_gemm_wmma<true><<<grid, 128, 0, stream>>>(buf, W1, b1, h, CAP, DFF, DMODEL);
    }
    {   // y = h @ W2 + b2           M=320, N=512, K=2048
        const int grid = NEXP * (CAP / 64) * (DMODEL / 64);
        moe_gemm_wmma<false><<<grid, 128, 0, stream>>>(h, W2, b2, yb, CAP, DMODEL, DFF);
    }

    combine_kernel<<<TOKENS, 128, 0, stream>>>(yb, pairSlot, out);
}